// GraphSAGE_61460982006090
// MI455X (gfx1250) — compile-verified
//
#include <hip/hip_runtime.h>

#define D 128
#define NNODES 50000
#define LDS_STRIDE 132   // 128 + 4 pad: conflict-free ds_load_b64 across both lane halves

typedef float v2f __attribute__((ext_vector_type(2)));
typedef float v8f __attribute__((ext_vector_type(8)));

// ---------------------------------------------------------------------------
// Kernel 1: zero the scatter workspace (sums + degrees)
// ---------------------------------------------------------------------------
__global__ void sage_zero_kernel(float* __restrict__ p, long n) {
    long i = (long)blockIdx.x * blockDim.x + threadIdx.x;
    long stride = (long)gridDim.x * blockDim.x;
    for (; i < n; i += stride) p[i] = 0.0f;
}

// ---------------------------------------------------------------------------
// Kernel 2: warp-per-edge scatter-add.  lane l handles dims [4l, 4l+4).
// src/dst ids are int64 (reference uses jnp.int64).
// ---------------------------------------------------------------------------
__global__ void __launch_bounds__(256)
sage_scatter_kernel(const float* __restrict__ src_feat,
                    const long long* __restrict__ src_ids,
                    const long long* __restrict__ dst_ids,
                    float* __restrict__ sum,
                    float* __restrict__ deg,
                    int n_edges) {
    int edge = (int)((blockIdx.x * (unsigned)blockDim.x + threadIdx.x) >> 5);
    int lane = threadIdx.x & 31;
    if (edge >= n_edges) return;

    int s = (int)src_ids[edge];   // uniform per wave
    int d = (int)dst_ids[edge];

    float4 v = *(const float4*)(src_feat + (long)s * D + lane * 4);
    float* o = sum + (long)d * D + lane * 4;
    atomicAdd(o + 0, v.x);
    atomicAdd(o + 1, v.y);
    atomicAdd(o + 2, v.z);
    atomicAdd(o + 3, v.w);
    if (lane == 0) atomicAdd(deg + d, 1.0f);
}

// ---------------------------------------------------------------------------
// Kernel 3: fused mean-normalize + dual GEMM (WMMA f32 16x16x4) + ReLU.
// Block = 256 threads = 8 waves; block owns 16 nodes; wave w owns output
// columns [16w, 16w+16).  out[n,j] = relu( sum_k self[n,k]*W_self[j,k]
//                                        + sum_k neigh[n,k]*W_neigh[j,k] )
// ---------------------------------------------------------------------------
__global__ void __launch_bounds__(256, 2)
sage_gemm_kernel(const float* __restrict__ self_feat,
                 const float* __restrict__ neigh_sum,
                 const float* __restrict__ deg,
                 const float* __restrict__ W_self,
                 const float* __restrict__ W_neigh,
                 float* __restrict__ out) {
    __shared__ float s_self[16 * LDS_STRIDE];
    __shared__ float s_neigh[16 * LDS_STRIDE];

    const int tid = threadIdx.x;
    const int node_base = blockIdx.x * 16;

    // Stage 16x128 self tile and degree-normalized neighbor tile into LDS.
    // 512 float4 slots, 2 per thread.
    for (int i = tid; i < 512; i += 256) {
        int row = i >> 5;
        int c4  = (i & 31) << 2;
        int node = node_base + row;
        float4 sv = *(const float4*)(self_feat + (long)node * D + c4);
        float4 nv = *(const float4*)(neigh_sum + (long)node * D + c4);
        float inv = 1.0f / fmaxf(deg[node], 1.0f);   // DGL mean semantics
        nv.x *= inv; nv.y *= inv; nv.z *= inv; nv.w *= inv;
        *(float4*)(s_self  + row * LDS_STRIDE + c4) = sv;
        *(float4*)(s_neigh + row * LDS_STRIDE + c4) = nv;
    }
    __syncthreads();

    const int wave  = tid >> 5;
    const int lane  = tid & 31;
    const int jbase = wave * 16;
    const int half  = lane >> 4;     // 0: K=0,1  |  1: K=2,3 (A frag layout)
    const int lrow  = lane & 15;     // A: matrix row M; B: output column N
    const int koff  = half * 2;

    // B = W^T, so lane's float2 is a contiguous pair within W row (jbase+lrow)
    const float* wsrow = W_self  + (jbase + lrow) * D + koff;
    const float* wnrow = W_neigh + (jbase + lrow) * D + koff;
    const float* asrow = s_self  + lrow * LDS_STRIDE + koff;
    const float* anrow = s_neigh + lrow * LDS_STRIDE + koff;

    v8f c = {0.f, 0.f, 0.f, 0.f, 0.f, 0.f, 0.f, 0.f};

#pragma unroll
    for (int k0 = 0; k0 < D; k0 += 4) {
        v2f a_s = *(const v2f*)(asrow + k0);
        v2f b_s = *(const v2f*)(wsrow + k0);
        c = __builtin_amdgcn_wmma_f32_16x16x4_f32(false, a_s, false, b_s,
                                                  (short)0, c, false, false);
        v2f a_n = *(const v2f*)(anrow + k0);
        v2f b_n = *(const v2f*)(wnrow + k0);
        c = __builtin_amdgcn_wmma_f32_16x16x4_f32(false, a_n, false, b_n,
                                                  (short)0, c, false, false);
    }

    // C/D layout: VGPR r = row M=r (lanes 0-15) / M=r+8 (lanes 16-31), N=lane&15
    const int col = jbase + lrow;
#pragma unroll
    for (int r = 0; r < 8; ++r) {
        int node = node_base + r + half * 8;
        float v = c[r];
        out[(long)node * D + col] = v > 0.0f ? v : 0.0f;
    }
}

// ---------------------------------------------------------------------------
extern "C" void kernel_launch(void* const* d_in, const int* in_sizes, int n_in,
                              void* d_out, int out_size, void* d_ws, size_t ws_size,
                              hipStream_t stream) {
    const float* user_embs      = (const float*)d_in[0];
    const float* item_embs      = (const float*)d_in[1];
    const float* W_self         = (const float*)d_in[2];
    const float* W_neigh        = (const float*)d_in[3];
    const long long* edges_iu   = (const long long*)d_in[4];  // [2, E] int64
    const long long* edges_ui   = (const long long*)d_in[5];
    const int E = in_sizes[4] / 2;

    // Workspace layout
    float* sum_u = (float*)d_ws;                    // [NNODES, D]
    float* sum_i = sum_u + (long)NNODES * D;        // [NNODES, D]
    float* deg_u = sum_i + (long)NNODES * D;        // [NNODES]
    float* deg_i = deg_u + NNODES;                  // [NNODES]

    long ztotal = 2L * NNODES * D + 2L * NNODES;
    sage_zero_kernel<<<2048, 256, 0, stream>>>((float*)d_ws, ztotal);

    int scatter_blocks = (E + 7) / 8;   // 8 edges per 256-thread block
    // item -> user aggregation
    sage_scatter_kernel<<<scatter_blocks, 256, 0, stream>>>(
        item_embs, edges_iu, edges_iu + E, sum_u, deg_u, E);
    // user -> item aggregation
    sage_scatter_kernel<<<scatter_blocks, 256, 0, stream>>>(
        user_embs, edges_ui, edges_ui + E, sum_i, deg_i, E);

    float* out = (float*)d_out;   // [2, NNODES, D]: users then items
    sage_gemm_kernel<<<NNODES / 16, 256, 0, stream>>>(
        user_embs, sum_u, deg_u, W_self, W_neigh, out);
    sage_gemm_kernel<<<NNODES / 16, 256, 0, stream>>>(
        item_embs, sum_i, deg_i, W_self, W_neigh, out + (long)NNODES * D);
}